// CapsuleLayer_31980326486217
// MI455X (gfx1250) — compile-verified
//
#include <hip/hip_runtime.h>

// Capsule routing for MI455X (gfx1250, wave32).
// Phase 1: priors[c,b,n,o] = sum_i x[b,n,i] * W[c,n,i,o] via V_WMMA_F32_16X16X4_F32.
// Phase 2: 3 dynamic-routing iterations, priors slice register-resident per workgroup.

typedef float v2f __attribute__((ext_vector_type(2)));
typedef float v8f __attribute__((ext_vector_type(8)));

#define C_CAPS  32
#define B_SZ    16
#define N_NODES 1024
#define CI      64
#define O_SZ    32

// x:      [B][N][Ci]      -> b*(N*Ci) + n*Ci + i
// W:      [C][N][Ci][O]   -> c*(N*Ci*O) + n*(Ci*O) + i*O + o
// priors: [C][B][N][O]    -> ((c*B + b)*N + n)*O + o        (in d_ws, 64 MB)

// ---------------------------------------------------------------------------
// Phase 1: one wave per (c, n). 16x16x4 f32 WMMA, K=64 in 16 steps, 2 o-tiles.
//
// VGPR layout assumptions (CDNA5 ISA 7.12.2):
//   A (16x4 f32): lanes 0-15 hold M=lane, K={0,1} in v0,v1; lanes 16-31 K={2,3}.
//   B (4x16 f32): mirrored — lane half selects K-pair, lane%16 = N column.
//   D (16x16 f32): VGPR r: lanes 0-15 -> M=r, N=lane; lanes 16-31 -> M=r+8.
// ---------------------------------------------------------------------------
__global__ __launch_bounds__(128) void caps_priors_wmma(
    const float* __restrict__ x, const float* __restrict__ W,
    float* __restrict__ priors)
{
    const int wid  = threadIdx.x >> 5;
    const int lane = threadIdx.x & 31;
    const int blk  = blockIdx.x;            // 0 .. C*(N/4)-1 = 8191
    const int c    = blk >> 8;              // 256 blocks per capsule class
    const int n    = ((blk & 255) << 2) + wid;

    const int m = lane & 15;                // A row (b) / B,D column (o base)
    const int h = lane >> 4;                // K-half selector

    const float* xp = x + (size_t)m * (N_NODES * CI) + (size_t)n * CI;
    const float* wp = W + (size_t)c * ((size_t)N_NODES * CI * O_SZ)
                        + (size_t)n * (CI * O_SZ);

    // Stream hint: prefetch the weight tile 4 n ahead (next block's work).
    __builtin_prefetch(wp + 4 * (CI * O_SZ), 0, 1);

    v8f acc0 = {};  // o = 0..15
    v8f acc1 = {};  // o = 16..31

    #pragma unroll
    for (int ks = 0; ks < 16; ++ks) {
        const int i0 = ks * 4 + 2 * h;      // this lane's K pair within the k-step

        v2f a;
        a.x = xp[i0];
        a.y = xp[i0 + 1];

        v2f b0, b1;
        b0.x = wp[(i0    ) * O_SZ + m     ];
        b0.y = wp[(i0 + 1) * O_SZ + m     ];
        b1.x = wp[(i0    ) * O_SZ + m + 16];
        b1.y = wp[(i0 + 1) * O_SZ + m + 16];

        acc0 = __builtin_amdgcn_wmma_f32_16x16x4_f32(
                   false, a, false, b0, (short)0, acc0, false, false);
        acc1 = __builtin_amdgcn_wmma_f32_16x16x4_f32(
                   false, a, false, b1, (short)0, acc1, false, false);
    }

    // Scatter D per the 16x16 f32 C/D layout. Half-wave stores are 64B-contiguous in o.
    float* pp = priors + ((size_t)c * B_SZ * N_NODES + (size_t)n) * O_SZ;
    #pragma unroll
    for (int r = 0; r < 8; ++r) {
        const int brow = r + 8 * h;
        pp[(size_t)brow * (N_NODES * O_SZ) + m     ] = acc0[r];
        pp[(size_t)brow * (N_NODES * O_SZ) + m + 16] = acc1[r];
    }
}

// ---------------------------------------------------------------------------
// Phase 2: one 256-thread workgroup per (c,b). Thread t owns rows n = t + 256*j,
// j=0..3, kept entirely in registers (4x32 floats). Deterministic tree reductions.
// ---------------------------------------------------------------------------
__global__ __launch_bounds__(256) void caps_route(
    const float* __restrict__ priors, float* __restrict__ out)
{
    const int cb   = blockIdx.x;            // c*B + b, 0..511
    const int t    = threadIdx.x;
    const int lane = t & 31;
    const int wid  = t >> 5;

    __shared__ float redv[8 * O_SZ];        // per-wave partial s[o]
    __shared__ float reds[8];               // per-wave scalar partials

    // Load this thread's 4 priors rows (128 B each, contiguous) into registers.
    float P[4][O_SZ];
    const float* base = priors + (size_t)cb * (N_NODES * O_SZ);
    #pragma unroll
    for (int j = 0; j < 4; ++j) {
        const float4* row = (const float4*)(base + (size_t)(t + 256 * j) * O_SZ);
        #pragma unroll
        for (int q = 0; q < 8; ++q) {
            float4 v = row[q];
            P[j][4*q+0] = v.x; P[j][4*q+1] = v.y;
            P[j][4*q+2] = v.z; P[j][4*q+3] = v.w;
        }
    }

    float logits[4] = {0.f, 0.f, 0.f, 0.f};
    float vout[O_SZ];

    for (int it = 0; it < 3; ++it) {
        // --- softmax max over all n ---
        float mx = fmaxf(fmaxf(logits[0], logits[1]), fmaxf(logits[2], logits[3]));
        #pragma unroll
        for (int off = 16; off; off >>= 1)
            mx = fmaxf(mx, __shfl_xor(mx, off, 32));
        if (lane == 0) reds[wid] = mx;
        __syncthreads();
        mx = reds[0];
        #pragma unroll
        for (int w = 1; w < 8; ++w) mx = fmaxf(mx, reds[w]);
        __syncthreads();

        // --- exp, partial Z, partial weighted sum s[o] ---
        float e[4]; float zp = 0.f;
        #pragma unroll
        for (int j = 0; j < 4; ++j) { e[j] = __expf(logits[j] - mx); zp += e[j]; }
        #pragma unroll
        for (int off = 16; off; off >>= 1) zp += __shfl_xor(zp, off, 32);
        if (lane == 0) reds[wid] = zp;

        #pragma unroll
        for (int o = 0; o < O_SZ; ++o) {
            float acc = e[0]*P[0][o] + e[1]*P[1][o] + e[2]*P[2][o] + e[3]*P[3][o];
            #pragma unroll
            for (int off = 16; off; off >>= 1) acc += __shfl_xor(acc, off, 32);
            if (lane == 0) redv[wid * O_SZ + o] = acc;
        }
        __syncthreads();

        // --- finalize s[o], squash ---
        float Z = 0.f;
        #pragma unroll
        for (int w = 0; w < 8; ++w) Z += reds[w];

        float sq = 0.f;
        #pragma unroll
        for (int o = 0; o < O_SZ; ++o) {
            float so = 0.f;
            #pragma unroll
            for (int w = 0; w < 8; ++w) so += redv[w * O_SZ + o];
            so /= Z;
            vout[o] = so;
            sq += so * so;
        }
        const float rs = sqrtf(sq);
        const float k  = rs / (1.0f + sq);   // squash: t * sqrt(sq)/(1+sq)
        #pragma unroll
        for (int o = 0; o < O_SZ; ++o) vout[o] *= k;

        // --- logit update (thread-local, priors already in registers) ---
        if (it < 2) {
            #pragma unroll
            for (int j = 0; j < 4; ++j) {
                float d = 0.f;
                #pragma unroll
                for (int o = 0; o < O_SZ; ++o) d += P[j][o] * vout[o];
                logits[j] += d;
            }
        }
        __syncthreads();
    }

    if (t == 0) {
        float* op = out + (size_t)cb * O_SZ;
        #pragma unroll
        for (int o = 0; o < O_SZ; ++o) op[o] = vout[o];
    }
}

extern "C" void kernel_launch(void* const* d_in, const int* in_sizes, int n_in,
                              void* d_out, int out_size, void* d_ws, size_t ws_size,
                              hipStream_t stream) {
    const float* x = (const float*)d_in[0];           // (16, 1024, 64) fp32
    const float* W = (const float*)d_in[1];           // (32, 1024, 64, 32) fp32
    float* priors  = (float*)d_ws;                    // (32, 16, 1024, 32) = 64 MB
    float* out     = (float*)d_out;                   // (32, 16, 1, 1, 32)

    // Phase 1: C * N/4 blocks of 4 waves; one wave per (c, n).
    caps_priors_wmma<<<dim3(C_CAPS * (N_NODES / 4)), dim3(128), 0, stream>>>(x, W, priors);
    // Phase 2: one workgroup per (c, b).
    caps_route<<<dim3(C_CAPS * B_SZ), dim3(256), 0, stream>>>(priors, out);
}